// LstmEprop1AllTimesteps_48430051229760
// MI455X (gfx1250) — compile-verified
//
#include <hip/hip_runtime.h>
#include <cstddef>

// Problem constants
constexpr int B = 16, T = 128, I = 64, H = 128, O = 32;
constexpr int G4H = 4 * H;                 // 512
constexpr int HP  = H + 1;                 // padded LDS row stride (bank-conflict-free WMMA A reads)
constexpr size_t NS = (size_t)T * B * H;   // 262144 elements per (t,b,h) slab

// ---- d_out layout (flattened return tuple, in floats) ----
constexpr size_t OFF_YS    = 0;                                   // (T*B, O)
constexpr size_t OFF_OUTS  = OFF_YS    + (size_t)T * B * O;       // (T,B,H)
constexpr size_t OFF_HT    = OFF_OUTS  + NS;                      // (B,H)
constexpr size_t OFF_CT    = OFF_HT    + (size_t)B * H;           // (B,H)
constexpr size_t OFF_ETX   = OFF_CT    + (size_t)B * H;           // (T,B,4H,I)
constexpr size_t OFF_ETH   = OFF_ETX   + (size_t)T * B * G4H * I; // (T,B,4H,H)
constexpr size_t OFF_ETB0  = OFF_ETH   + (size_t)T * B * G4H * H; // (T,B,4H)
constexpr size_t OFF_ETB1  = OFF_ETB0  + (size_t)T * B * G4H;     // (T,B,4H)
constexpr size_t OFF_GATES = OFF_ETB1  + (size_t)T * B * G4H;     // (T,B,4H)
constexpr size_t OFF_CSEQ  = OFF_GATES + (size_t)T * B * G4H;     // (T,B,H)
constexpr size_t OFF_CPREV = OFF_CSEQ  + NS;                      // (T,B,H)
constexpr size_t OFF_HPREV = OFF_CPREV + NS;                      // (T,B,H)

// ---- workspace layout (floats): 6 per-(t,b,h) scalar slabs + pre_x ----
// [0]=f, [1]=inst_i, [2]=inst_f, [3]=inst_g, [4]=psi, [5]=do_t
constexpr size_t WS_PREX = 6 * NS;                                // (T*B, 4H)
// total ws use = 6*NS + T*B*4H floats = 10.5 MB

typedef __attribute__((ext_vector_type(2))) float v2f;
typedef __attribute__((ext_vector_type(8))) float v8f;

__device__ __forceinline__ v8f wmma_f32_k4(v2f a, v2f b, v8f c) {
  // D = A(16x4 f32) x B(4x16 f32) + C(16x16 f32)
  return __builtin_amdgcn_wmma_f32_16x16x4_f32(false, a, false, b, (short)0, c,
                                               false, false);
}

__device__ __forceinline__ float sigmoidf_(float x) {
  return 1.0f / (1.0f + __expf(-x));
}

// ---------------------------------------------------------------------------
// Kernel 0: pre_x[t*B+b, r] = x_t . W_ih^T + (b_ih + b_hh), all timesteps.
// M = T*B = 2048 (Mtile == t exactly since B==16), N = 512, K = 64.
// One 16x16 tile per wave; 8 waves per block; 4096 tile jobs -> 512 blocks.
// ---------------------------------------------------------------------------
__global__ __launch_bounds__(256) void eprop_prex_kernel(
    const float* __restrict__ inp, const float* __restrict__ W_ih,
    const float* __restrict__ b_ih, const float* __restrict__ b_hh,
    float* __restrict__ ws) {
  const int lane = threadIdx.x & 31;
  const int half = lane >> 4;
  const int lr   = lane & 15;
  const int job  = blockIdx.x * 8 + (threadIdx.x >> 5);  // 0..4095
  const int mt   = job >> 5;   // == t
  const int nt   = job & 31;
  const int r    = nt * 16 + lr;

  const float* xrow = inp + (size_t)lr * T * I + (size_t)mt * I;  // x[b=lr][t=mt][*]
  v8f acc = {};
  #pragma unroll
  for (int kb = 0; kb < I; kb += 4) {
    v2f a, bm;
    a.x  = xrow[kb + 2 * half + 0];
    a.y  = xrow[kb + 2 * half + 1];
    bm.x = W_ih[(size_t)r * I + kb + 2 * half + 0];
    bm.y = W_ih[(size_t)r * I + kb + 2 * half + 1];
    acc = wmma_f32_k4(a, bm, acc);
  }
  const float bias = b_ih[r] + b_hh[r];
  #pragma unroll
  for (int v = 0; v < 8; ++v) {
    const int m = mt * 16 + half * 8 + v;  // = t*B + b
    ws[WS_PREX + (size_t)m * G4H + r] = acc[v] + bias;
  }
}

// ---------------------------------------------------------------------------
// Kernel 1: sequential LSTM recurrence. One workgroup, 512 threads = 16 waves.
// W_hh (256 KB) is staged ONCE into LDS, transposed to k-major so B-fragment
// ds_loads are bank-conflict-free (lane stride = 1 word). s_h rows padded to
// H+1 so A-fragment reads (lane stride 129 words) avoid 16-way conflicts.
// Total LDS: 256K (W_hh) + 32K (pre) + ~8.3K (h) + ~8.3K (c) = ~304 KB of the
// 320 KB WGP budget.
// ---------------------------------------------------------------------------
__global__ __launch_bounds__(512) void eprop_recur_kernel(
    const float* __restrict__ W_hh, float* __restrict__ out,
    float* __restrict__ ws) {
  __shared__ float s_whh[H * G4H];  // k-major: s_whh[k*512 + r], 256 KB
  __shared__ float s_pre[B * G4H];  // 32 KB
  __shared__ float s_h[B * HP];     // padded, ~8.3 KB
  __shared__ float s_c[B * HP];     // padded, ~8.3 KB

  const int tid  = threadIdx.x;
  const int wave = tid >> 5;       // 0..15
  const int lane = tid & 31;
  const int half = lane >> 4;
  const int lr   = lane & 15;

  // one-time: stage W_hh transposed into LDS; init h, c
  for (int idx = tid; idx < H * G4H; idx += 512) {
    const int r = idx >> 7;        // row of W_hh (gate column), 0..511
    const int k = idx & 127;       // k index, 0..127
    s_whh[k * G4H + r] = W_hh[idx];
  }
  for (int e = tid; e < B * HP; e += 512) { s_h[e] = 0.0f; s_c[e] = 0.0f; }
  __syncthreads();

  for (int t = 0; t < T; ++t) {
    // ---- phase 1: pre = pre_x + h . W_hh^T, via WMMA (LDS-fed) ----
    for (int nt = 0; nt < 2; ++nt) {
      const int n = wave + nt * 16;  // N-tile 0..31
      const int r = n * 16 + lr;     // gate column 0..511
      v8f acc;
      #pragma unroll
      for (int v = 0; v < 8; ++v)
        acc[v] = ws[WS_PREX + ((size_t)t * 16 + half * 8 + v) * G4H + r];
      #pragma unroll
      for (int kb = 0; kb < H; kb += 4) {
        v2f a, bm;
        a.x  = s_h[lr * HP + kb + 2 * half + 0];
        a.y  = s_h[lr * HP + kb + 2 * half + 1];
        bm.x = s_whh[(kb + 2 * half + 0) * G4H + r];
        bm.y = s_whh[(kb + 2 * half + 1) * G4H + r];
        acc = wmma_f32_k4(a, bm, acc);
      }
      #pragma unroll
      for (int v = 0; v < 8; ++v)
        s_pre[(half * 8 + v) * G4H + r] = acc[v];
    }
    __syncthreads();

    // ---- phase 2: gates, cell update, caches, trace scalars ----
    for (int e = tid; e < B * H; e += 512) {
      const int b  = e >> 7;
      const int hc = e & 127;
      const int ep = b * HP + hc;    // padded LDS index
      const float pi = s_pre[b * G4H + hc];
      const float pf = s_pre[b * G4H + H + hc];
      const float pg = s_pre[b * G4H + 2 * H + hc];
      const float po = s_pre[b * G4H + 3 * H + hc];
      const float i_ = sigmoidf_(pi);
      const float f_ = sigmoidf_(pf);
      const float g_ = tanhf(pg);
      const float o_ = sigmoidf_(po);
      const float c_prev = s_c[ep];
      const float h_prev = s_h[ep];
      const float c_new  = f_ * c_prev + i_ * g_;
      const float tc     = tanhf(c_new);
      const float h_new  = o_ * tc;

      const size_t tb = (size_t)t * B * H + e;  // [t][b][hc]
      out[OFF_OUTS  + tb] = h_new;
      out[OFF_CSEQ  + tb] = c_new;
      out[OFF_CPREV + tb] = c_prev;
      out[OFF_HPREV + tb] = h_prev;
      const size_t gb = OFF_GATES + (size_t)t * B * G4H + (size_t)b * G4H;
      out[gb + hc]         = i_;
      out[gb + H + hc]     = f_;
      out[gb + 2 * H + hc] = g_;
      out[gb + 3 * H + hc] = o_;

      ws[0 * NS + tb] = f_;
      ws[1 * NS + tb] = i_ * (1.0f - i_) * g_;       // inst_i
      ws[2 * NS + tb] = f_ * (1.0f - f_) * c_prev;   // inst_f
      ws[3 * NS + tb] = (1.0f - g_ * g_) * i_;       // inst_g
      ws[4 * NS + tb] = o_ * (1.0f - tc * tc);       // psi
      ws[5 * NS + tb] = tc * o_ * (1.0f - o_);       // do_t

      s_c[ep] = c_new;
      s_h[ep] = h_new;
      if (t == T - 1) { out[OFF_HT + e] = h_new; out[OFF_CT + e] = c_new; }
    }
    __syncthreads();
  }
}

// ---------------------------------------------------------------------------
// Kernel 2: et_wih. One thread per (b, d, h, col) chain; private ev recurrence
// over t; 1 KB contiguous non-temporal stores per block per t (pure stream,
// never re-read -> keep it out of L2). rows = B*4H = 8192, cols = 64.
// ---------------------------------------------------------------------------
__global__ __launch_bounds__(256) void eprop_etx_kernel(
    const float* __restrict__ inp, const float* __restrict__ ws,
    float* __restrict__ out) {
  const int col = threadIdx.x & 63;
  const int g   = blockIdx.x * 4 + (threadIdx.x >> 6);  // 0..8191
  const int b   = g >> 9;
  const int row = g & 511;
  const int d   = row >> 7;
  const int h   = row & 127;

  const float* xp = inp + (size_t)b * T * I + col;   // stride I per t
  const size_t bh = (size_t)b * H + h;
  const size_t st = (size_t)B * H;                   // t-stride in ws slabs

  if (d < 3) {
    const float* fp = ws + 0 * NS + bh;
    const float* ip = ws + (size_t)(1 + d) * NS + bh;
    const float* pp = ws + 4 * NS + bh;
    float ev = 0.0f;
    for (int t = 0; t < T; ++t) {
      const float x = xp[(size_t)t * I];
      ev = fp[t * st] * ev + ip[t * st] * x;
      __builtin_nontemporal_store(
          pp[t * st] * ev,
          &out[OFF_ETX + (((size_t)t * B + b) * G4H + row) * I + col]);
    }
  } else {
    const float* dp = ws + 5 * NS + bh;
    for (int t = 0; t < T; ++t) {
      const float x = xp[(size_t)t * I];
      __builtin_nontemporal_store(
          dp[t * st] * x,
          &out[OFF_ETX + (((size_t)t * B + b) * G4H + row) * I + col]);
    }
  }
}

// ---------------------------------------------------------------------------
// Kernel 3: et_whh. rows = B*4H = 8192, cols = H = 128. 2 rows per block.
// h_prev streamed (RT) from the cached h_prev_seq region; output stores NT.
// ---------------------------------------------------------------------------
__global__ __launch_bounds__(256) void eprop_eth_kernel(
    const float* __restrict__ ws, float* __restrict__ out) {
  const int col = threadIdx.x & 127;
  const int g   = blockIdx.x * 2 + (threadIdx.x >> 7);  // 0..8191
  const int b   = g >> 9;
  const int row = g & 511;
  const int d   = row >> 7;
  const int h   = row & 127;

  const float* hp = out + OFF_HPREV + (size_t)b * H + col;  // stride B*H per t
  const size_t bh = (size_t)b * H + h;
  const size_t st = (size_t)B * H;

  if (d < 3) {
    const float* fp = ws + 0 * NS + bh;
    const float* ip = ws + (size_t)(1 + d) * NS + bh;
    const float* pp = ws + 4 * NS + bh;
    float ev = 0.0f;
    for (int t = 0; t < T; ++t) {
      const float hv = hp[t * st];
      ev = fp[t * st] * ev + ip[t * st] * hv;
      __builtin_nontemporal_store(
          pp[t * st] * ev,
          &out[OFF_ETH + (((size_t)t * B + b) * G4H + row) * H + col]);
    }
  } else {
    const float* dp = ws + 5 * NS + bh;
    for (int t = 0; t < T; ++t) {
      const float hv = hp[t * st];
      __builtin_nontemporal_store(
          dp[t * st] * hv,
          &out[OFF_ETH + (((size_t)t * B + b) * G4H + row) * H + col]);
    }
  }
}

// ---------------------------------------------------------------------------
// Kernel 4: et_b (written to both duplicate output regions, NT stores).
// 8192 chains, one thread each.
// ---------------------------------------------------------------------------
__global__ __launch_bounds__(256) void eprop_etb_kernel(
    const float* __restrict__ ws, float* __restrict__ out) {
  const int g   = blockIdx.x * 256 + threadIdx.x;  // 0..8191
  const int b   = g >> 9;
  const int row = g & 511;
  const int d   = row >> 7;
  const size_t bh = (size_t)b * H + (row & 127);
  const size_t st = (size_t)B * H;

  float ev = 0.0f;
  for (int t = 0; t < T; ++t) {
    float val;
    if (d < 3) {
      ev = ws[0 * NS + bh + t * st] * ev + ws[(size_t)(1 + d) * NS + bh + t * st];
      val = ws[4 * NS + bh + t * st] * ev;
    } else {
      val = ws[5 * NS + bh + t * st];
    }
    const size_t o = ((size_t)t * B + b) * G4H + row;
    __builtin_nontemporal_store(val, &out[OFF_ETB0 + o]);
    __builtin_nontemporal_store(val, &out[OFF_ETB1 + o]);
  }
}

// ---------------------------------------------------------------------------
// Kernel 5: ys = outs.reshape(T*B, H) @ W_out^T.  M=2048, N=32, K=128.
// 128 Mtiles x 2 Ntiles = 256 wave-jobs; 8 waves/block -> 32 blocks.
// ---------------------------------------------------------------------------
__global__ __launch_bounds__(256) void eprop_ys_kernel(
    const float* __restrict__ W_out, float* __restrict__ out) {
  const int lane = threadIdx.x & 31;
  const int half = lane >> 4;
  const int lr   = lane & 15;
  const int job  = blockIdx.x * 8 + (threadIdx.x >> 5);  // 0..255
  const int mt   = job >> 1;
  const int nt   = job & 1;

  const float* A = out + OFF_OUTS;   // (2048, 128) row-major
  const int m = mt * 16 + lr;
  const int n = nt * 16 + lr;

  v8f acc = {};
  #pragma unroll
  for (int kb = 0; kb < H; kb += 4) {
    v2f a, bm;
    a.x  = A[(size_t)m * H + kb + 2 * half + 0];
    a.y  = A[(size_t)m * H + kb + 2 * half + 1];
    bm.x = W_out[(size_t)n * H + kb + 2 * half + 0];
    bm.y = W_out[(size_t)n * H + kb + 2 * half + 1];
    acc = wmma_f32_k4(a, bm, acc);
  }
  #pragma unroll
  for (int v = 0; v < 8; ++v) {
    const int mr = mt * 16 + half * 8 + v;
    out[OFF_YS + (size_t)mr * O + nt * 16 + lr] = acc[v];
  }
}

// ---------------------------------------------------------------------------
extern "C" void kernel_launch(void* const* d_in, const int* in_sizes, int n_in,
                              void* d_out, int out_size, void* d_ws,
                              size_t ws_size, hipStream_t stream) {
  const float* inp   = (const float*)d_in[0];  // (B,T,I)
  const float* W_ih  = (const float*)d_in[1];  // (4H,I)
  const float* W_hh  = (const float*)d_in[2];  // (4H,H)
  const float* b_ih  = (const float*)d_in[3];  // (4H,)
  const float* b_hh  = (const float*)d_in[4];  // (4H,)
  const float* W_out = (const float*)d_in[5];  // (O,H)
  float* out = (float*)d_out;
  float* ws  = (float*)d_ws;  // needs 6*NS + T*B*4H floats = 10.5 MB

  // 0) input-path GEMM for all timesteps (parallel, WMMA)
  eprop_prex_kernel<<<512, 256, 0, stream>>>(inp, W_ih, b_ih, b_hh, ws);
  // 1) sequential recurrence (WMMA on h.W_hh^T, LDS-resident weights)
  eprop_recur_kernel<<<1, 512, 0, stream>>>(W_hh, out, ws);
  // 2-4) eligibility-trace streaming (HBM-store bound, ~800 MB, NT stores)
  eprop_etx_kernel<<<2048, 256, 0, stream>>>(inp, ws, out);
  eprop_eth_kernel<<<4096, 256, 0, stream>>>(ws, out);
  eprop_etb_kernel<<<32, 256, 0, stream>>>(ws, out);
  // 5) output projection (WMMA)
  eprop_ys_kernel<<<32, 256, 0, stream>>>(W_out, out);
}